// RPN_86517821213526
// MI455X (gfx1250) — compile-verified
//
#include <hip/hip_runtime.h>
#include <hip/hip_bf16.h>

#define H_DIM 160
#define W_DIM 160
#define A_DIM 9
#define N_ANCH 230400            // H*W*A
#define K_GT 128
#define RPN_NEG 0.3f
#define RPN_POS 0.7f
#define RPN_BATCH 256
#define RPN_NUM_FG 128           // 0.5 * RPN_BATCH
#define NBUCKET 65536            // 16-bit histogram levels

// ---------------- CDNA5 async global->LDS staging (ASYNCcnt path) ----------
#if defined(__has_builtin)
#if __has_builtin(__builtin_amdgcn_global_load_async_to_lds_b128)
#define HAVE_ASYNC_BUILTIN 1
#endif
#if __has_builtin(__builtin_amdgcn_s_wait_asynccnt)
#define HAVE_WAIT_BUILTIN 1
#endif
#endif

typedef int v4i_t __attribute__((ext_vector_type(4)));
typedef __attribute__((address_space(1))) v4i_t v4i_global;
typedef __attribute__((address_space(3))) v4i_t v4i_lds;

__device__ __forceinline__ void async_copy16(void* lds_dst, const void* gsrc) {
#if defined(HAVE_ASYNC_BUILTIN)
  __builtin_amdgcn_global_load_async_to_lds_b128(
      (v4i_global*)gsrc, (v4i_lds*)lds_dst, 0, 0);
#else
  unsigned lo = (unsigned)(unsigned long long)
      (__attribute__((address_space(3))) void*)lds_dst;
  unsigned long long ga = (unsigned long long)gsrc;
  asm volatile("global_load_async_to_lds_b128 %0, %1, off"
               :: "v"(lo), "v"(ga) : "memory");
#endif
}

__device__ __forceinline__ void wait_async_zero() {
#if defined(HAVE_WAIT_BUILTIN)
  __builtin_amdgcn_s_wait_asynccnt(0);
#else
  asm volatile("s_wait_asynccnt 0" ::: "memory");
#endif
}

// ---------------- helpers ---------------------------------------------------
// Order-preserving float<->uint so atomicMax(u32) == exact float max.
__device__ __forceinline__ unsigned f2ord(float f) {
  unsigned b = __float_as_uint(f);
  return (b & 0x80000000u) ? ~b : (b | 0x80000000u);
}
__device__ __forceinline__ float ord2f(unsigned o) {
  unsigned b = (o & 0x80000000u) ? (o & 0x7FFFFFFFu) : ~o;
  return __uint_as_float(b);
}

// Shared by K1 and K2 so ov == gt_max equality is bit-reproducible.
__device__ __forceinline__ float iou_ov(float4 a, float4 g, float ga, bool inside) {
  float ba = (a.z - a.x + 1.0f) * (a.w - a.y + 1.0f);
  float iw = fminf(a.z, g.z) - fmaxf(a.x, g.x) + 1.0f;
  float ih = fminf(a.w, g.w) - fmaxf(a.y, g.y) + 1.0f;
  iw = fmaxf(iw, 0.0f);
  ih = fmaxf(ih, 0.0f);
  float inter = iw * ih;
  float ov = inter / (ba + ga - inter);
  return inside ? ov : -1.0f;
}

struct SelCtl {
  int      cnt_fg;  int apply_fg;  unsigned pref_fg;  int k_fg;  unsigned higher_fg;
  unsigned T_fg;    int kept_fg;
  int      cnt_bg;  int apply_bg;  unsigned pref_bg;  int k_bg;  unsigned higher_bg;
  unsigned T_bg;    int kept_bg;
  float    inv_num;
};

// ---------------- K0: init gt_max + zero 4x64K histograms -------------------
__global__ void rpn_k0_init(unsigned* __restrict__ gt_max_u,
                            unsigned* __restrict__ hists /* 4*NBUCKET */) {
  const int i = blockIdx.x * 256 + threadIdx.x;
  if (i < 4 * NBUCKET) hists[i] = 0u;
  if (i < K_GT) gt_max_u[i] = f2ord(-1.0f);
}

// ---------------- K1: per-gt column max over all anchors ---------------------
// Block = 256 threads (8 waves). 128 anchors per block staged in LDS (async).
#define K1_APB 128
__global__ void rpn_k1_gtmax(const float4* __restrict__ anchors,
                             const float4* __restrict__ gt,
                             const float* __restrict__ im_info,
                             unsigned* __restrict__ gt_max_u) {
  __shared__ float4 s_anch[K1_APB];
  __shared__ float s_red[K_GT];
  const int tid = threadIdx.x;
  const int base = blockIdx.x * K1_APB;

  if (tid < K1_APB) async_copy16(&s_anch[tid], &anchors[base + tid]);

  const int g = tid & (K_GT - 1);
  const int half = tid >> 7;
  const float4 gb = gt[g];
  const float ga = (gb.z - gb.x + 1.0f) * (gb.w - gb.y + 1.0f);
  const float im_h = im_info[0], im_w = im_info[1];

  wait_async_zero();
  __syncthreads();

  float m = -1.0f;
  const int a0 = half * (K1_APB / 2);
#pragma unroll 4
  for (int j = 0; j < K1_APB / 2; ++j) {
    float4 a = s_anch[a0 + j];  // LDS broadcast (same addr across lanes)
    bool inside = (a.x >= 0.0f) && (a.y >= 0.0f) && (a.z < im_w) && (a.w < im_h);
    m = fmaxf(m, iou_ov(a, gb, ga, inside));
  }
  if (half == 0) s_red[g] = m;
  __syncthreads();
  if (half == 1) {
    float mm = fmaxf(s_red[g], m);
    atomicMax(&gt_max_u[g], f2ord(mm));
  }
}

// ---------------- K2: labels (pre-subsample) + targets + MSB histograms -----
__global__ void rpn_k2_labels(const float4* __restrict__ anchors,
                              const float4* __restrict__ gt,
                              const float* __restrict__ im_info,
                              const unsigned* __restrict__ gt_max_u,
                              const float* __restrict__ noise,
                              float* __restrict__ labels_pre,
                              unsigned* __restrict__ msb_fg,
                              unsigned* __restrict__ msb_bg,
                              float* __restrict__ out) {
  __shared__ float4 s_gt[K_GT];
  __shared__ float s_gmax[K_GT];
  const int tid = threadIdx.x;
  const int i = blockIdx.x * 256 + tid;

  if (tid < K_GT) async_copy16(&s_gt[tid], &gt[tid]);
  if (tid < K_GT) s_gmax[tid] = ord2f(gt_max_u[tid]);

  float4 a = anchors[i];
  const float im_h = im_info[0], im_w = im_info[1];
  bool inside = (a.x >= 0.0f) && (a.y >= 0.0f) && (a.z < im_w) && (a.w < im_h);

  wait_async_zero();
  __syncthreads();

  float m = -__builtin_inff();
  int amax = 0;
  bool isbest = false;
#pragma unroll 4
  for (int g = 0; g < K_GT; ++g) {
    float4 gb = s_gt[g];
    float ga = (gb.z - gb.x + 1.0f) * (gb.w - gb.y + 1.0f);
    float ov = iou_ov(a, gb, ga, inside);
    isbest = isbest || (ov == s_gmax[g]);
    if (ov > m) { m = ov; amax = g; }  // strict > == jnp.argmax first-tie rule
  }
  isbest = isbest && inside;

  float l = -1.0f;
  if (inside && m < RPN_NEG) l = 0.0f;   // bg first (no clobber of later fg)
  if (isbest) l = 1.0f;
  if (inside && m >= RPN_POS) l = 1.0f;
  labels_pre[i] = l;

  // MSB (top-16-bit) noise histograms for the top-k selection
  if (l >= 0.0f) {
    unsigned b = __float_as_uint(noise[i]);
    unsigned* h = (l == 1.0f) ? msb_fg : msb_bg;
    atomicAdd(&h[b >> 16], 1u);
  }

  // bbox_transform against matched gt; zero for outside anchors (_unmap)
  float4 gb = s_gt[amax];
  float ew = a.z - a.x + 1.0f, eh = a.w - a.y + 1.0f;
  float ecx = a.x + 0.5f * ew, ecy = a.y + 0.5f * eh;
  float gw = gb.z - gb.x + 1.0f, gh = gb.w - gb.y + 1.0f;
  float gcx = gb.x + 0.5f * gw, gcy = gb.y + 0.5f * gh;
  float4 t;
  t.x = (gcx - ecx) / ew;
  t.y = (gcy - ecy) / eh;
  t.z = logf(gw / ew);
  t.w = logf(gh / eh);
  if (!inside) { t.x = 0.0f; t.y = 0.0f; t.z = 0.0f; t.w = 0.0f; }
  ((float4*)(out + N_ANCH))[i] = t;
}

// ---------------- K3a: fg MSB bucket select (1 block) ------------------------
__global__ void rpn_k3a_fg_msb(const unsigned* __restrict__ msb_fg,
                               SelCtl* __restrict__ ctl) {
  __shared__ unsigned csum[256];
  const int tid = threadIdx.x;
  unsigned s = 0;
  const int base = tid * 256;
  for (int j = 0; j < 256; ++j) s += msb_fg[base + j];
  csum[tid] = s;
  __syncthreads();
  if (tid == 0) {
    unsigned total = 0;
    for (int d = 0; d < 256; ++d) total += csum[d];
    const int cnt = (int)total;
    ctl->cnt_fg = cnt;
    const int apply = (cnt > RPN_NUM_FG) ? 1 : 0;
    ctl->apply_fg = apply;
    if (apply) {
      const unsigned k = RPN_NUM_FG;        // clamp(128,1,N) == 128
      unsigned cum = 0; int c = 255;
      for (; c > 0; --c) { if (cum + csum[c] >= k) break; cum += csum[c]; }
      unsigned higher = cum; int sel = 0;
      for (int j = 255; j >= 0; --j) {
        unsigned v = msb_fg[c * 256 + j];
        if (higher + v >= k) { sel = c * 256 + j; break; }
        higher += v;
      }
      ctl->pref_fg = (unsigned)sel;
      ctl->k_fg = (int)(k - higher);
      ctl->higher_fg = higher;
    } else {
      ctl->pref_fg = 0u; ctl->k_fg = 0; ctl->higher_fg = 0u;
      ctl->T_fg = 0u; ctl->kept_fg = cnt;   // finalized (no subsampling)
    }
  }
}

// ---------------- K3b/K3d: conditional low-16-bit histogram (grid-wide) -----
__global__ void rpn_k3_lowhist(const float* __restrict__ labels_pre,
                               const float* __restrict__ noise,
                               const SelCtl* __restrict__ ctl,
                               unsigned* __restrict__ low_hist,
                               int phase /* 0=fg, 1=bg */) {
  const int apply = (phase == 0) ? ctl->apply_fg : ctl->apply_bg;
  if (!apply) return;
  const unsigned pref = (phase == 0) ? ctl->pref_fg : ctl->pref_bg;
  const float maskv = (phase == 0) ? 1.0f : 0.0f;
  const int i = blockIdx.x * 256 + threadIdx.x;
  if (labels_pre[i] == maskv) {
    unsigned b = __float_as_uint(noise[i]);
    if ((b >> 16) == pref) atomicAdd(&low_hist[b & 0xFFFFu], 1u);
  }
}

// ---------------- K3c: fg low select + bg MSB select (1 block) ---------------
__global__ void rpn_k3c(const unsigned* __restrict__ low_fg,
                        const unsigned* __restrict__ msb_bg,
                        SelCtl* __restrict__ ctl) {
  __shared__ unsigned csum[256];
  const int tid = threadIdx.x;
  // ---- fg low-16 select: exact threshold + kept count from histogram sums --
  if (ctl->apply_fg) {
    unsigned s = 0;
    const int base = tid * 256;
    for (int j = 0; j < 256; ++j) s += low_fg[base + j];
    csum[tid] = s;
  }
  __syncthreads();
  if (tid == 0 && ctl->apply_fg) {
    const unsigned k = (unsigned)ctl->k_fg;
    unsigned cum = 0; int c = 255;
    for (; c > 0; --c) { if (cum + csum[c] >= k) break; cum += csum[c]; }
    unsigned higher = cum; int sel = 0; unsigned vat = 0;
    for (int j = 255; j >= 0; --j) {
      unsigned v = low_fg[c * 256 + j];
      if (higher + v >= k) { sel = c * 256 + j; vat = v; break; }
      higher += v;
    }
    ctl->T_fg = (ctl->pref_fg << 16) | (unsigned)sel;
    ctl->kept_fg = (int)(ctl->higher_fg + higher + vat); // #{bits >= T_fg}
  }
  __syncthreads();
  // ---- bg MSB select with num_bg = RPN_BATCH - kept_fg ---------------------
  {
    unsigned s = 0;
    const int base = tid * 256;
    for (int j = 0; j < 256; ++j) s += msb_bg[base + j];
    csum[tid] = s;
  }
  __syncthreads();
  if (tid == 0) {
    unsigned total = 0;
    for (int d = 0; d < 256; ++d) total += csum[d];
    const int cnt = (int)total;
    ctl->cnt_bg = cnt;
    const int num_bg = RPN_BATCH - ctl->kept_fg;
    const int apply = (cnt > num_bg) ? 1 : 0;
    ctl->apply_bg = apply;
    if (apply) {
      int kk = num_bg;                        // idx = clip(num_keep-1,0,N-1)
      if (kk < 1) kk = 1;
      if (kk > N_ANCH) kk = N_ANCH;
      const unsigned k = (unsigned)kk;
      unsigned cum = 0; int c = 255;
      for (; c > 0; --c) { if (cum + csum[c] >= k) break; cum += csum[c]; }
      unsigned higher = cum; int sel = 0;
      for (int j = 255; j >= 0; --j) {
        unsigned v = msb_bg[c * 256 + j];
        if (higher + v >= k) { sel = c * 256 + j; break; }
        higher += v;
      }
      ctl->pref_bg = (unsigned)sel;
      ctl->k_bg = (int)(k - higher);
      ctl->higher_bg = higher;
    } else {
      ctl->pref_bg = 0u; ctl->k_bg = 0; ctl->higher_bg = 0u;
      ctl->T_bg = 0u; ctl->kept_bg = cnt;   // finalized (no subsampling)
    }
  }
}

// ---------------- K3e: bg low select + num_examples (1 block) ----------------
__global__ void rpn_k3e(const unsigned* __restrict__ low_bg,
                        SelCtl* __restrict__ ctl) {
  __shared__ unsigned csum[256];
  const int tid = threadIdx.x;
  if (ctl->apply_bg) {
    unsigned s = 0;
    const int base = tid * 256;
    for (int j = 0; j < 256; ++j) s += low_bg[base + j];
    csum[tid] = s;
  }
  __syncthreads();
  if (tid == 0) {
    if (ctl->apply_bg) {
      const unsigned k = (unsigned)ctl->k_bg;
      unsigned cum = 0; int c = 255;
      for (; c > 0; --c) { if (cum + csum[c] >= k) break; cum += csum[c]; }
      unsigned higher = cum; int sel = 0; unsigned vat = 0;
      for (int j = 255; j >= 0; --j) {
        unsigned v = low_bg[c * 256 + j];
        if (higher + v >= k) { sel = c * 256 + j; vat = v; break; }
        higher += v;
      }
      ctl->T_bg = (ctl->pref_bg << 16) | (unsigned)sel;
      ctl->kept_bg = (int)(ctl->higher_bg + higher + vat);
    }
    const int num = ctl->kept_fg + ctl->kept_bg;  // sum(labels >= 0)
    ctl->inv_num = (num > 0) ? (1.0f / (float)num) : 0.0f;
  }
}

// ---------------- K4: final labels (permuted) + weights ----------------------
__global__ void rpn_k4_final(const float* __restrict__ labels_pre,
                             const float* __restrict__ noise,
                             const SelCtl* __restrict__ ctl,
                             float* __restrict__ out) {
  const int i = blockIdx.x * 256 + threadIdx.x;
  float l = labels_pre[i];
  const unsigned b = __float_as_uint(noise[i]);
  if (ctl->apply_fg && l == 1.0f && b < ctl->T_fg) l = -1.0f;
  if (ctl->apply_bg && l == 0.0f && b < ctl->T_bg) l = -1.0f;

  const int hw = i / A_DIM;           // h*W + w
  const int a = i - hw * A_DIM;
  out[a * (H_DIM * W_DIM) + hw] = l;  // (H,W,A) -> (A,H,W) flat

  const float iw = (l == 1.0f) ? 1.0f : 0.0f;
  const float ow = (l >= 0.0f) ? ctl->inv_num : 0.0f;
  float4 v;
  v.x = iw; v.y = iw; v.z = iw; v.w = iw;
  ((float4*)(out + 5 * N_ANCH))[i] = v;
  v.x = ow; v.y = ow; v.z = ow; v.w = ow;
  ((float4*)(out + 9 * N_ANCH))[i] = v;
}

// ---------------- launcher ---------------------------------------------------
extern "C" void kernel_launch(void* const* d_in, const int* in_sizes, int n_in,
                              void* d_out, int out_size, void* d_ws, size_t ws_size,
                              hipStream_t stream) {
  (void)in_sizes; (void)n_in; (void)out_size; (void)ws_size;
  const float* anchors = (const float*)d_in[0];   // [N,4]
  const float* gt      = (const float*)d_in[1];   // [K,4]
  const float* im_info = (const float*)d_in[2];   // [2]
  const float* noise   = (const float*)d_in[3];   // [N]
  float* out = (float*)d_out;                     // labels N | targets 4N | in_w 4N | out_w 4N

  unsigned char* ws = (unsigned char*)d_ws;
  size_t off = 0;
  unsigned* gt_max_u   = (unsigned*)(ws + off); off += 512;                       // 128*4, padded
  float*    labels_pre = (float*)(ws + off);    off += (size_t)N_ANCH * 4;        // 921600
  unsigned* hists      = (unsigned*)(ws + off); off += (size_t)4 * NBUCKET * 4;   // 1 MiB
  unsigned* msb_fg = hists;
  unsigned* msb_bg = hists + NBUCKET;
  unsigned* low_fg = hists + 2 * NBUCKET;
  unsigned* low_bg = hists + 3 * NBUCKET;
  SelCtl* ctl = (SelCtl*)(ws + off);

  rpn_k0_init<<<(4 * NBUCKET + 255) / 256, 256, 0, stream>>>(gt_max_u, hists);
  rpn_k1_gtmax<<<N_ANCH / K1_APB, 256, 0, stream>>>(
      (const float4*)anchors, (const float4*)gt, im_info, gt_max_u);
  rpn_k2_labels<<<N_ANCH / 256, 256, 0, stream>>>(
      (const float4*)anchors, (const float4*)gt, im_info, gt_max_u,
      noise, labels_pre, msb_fg, msb_bg, out);
  rpn_k3a_fg_msb<<<1, 256, 0, stream>>>(msb_fg, ctl);
  rpn_k3_lowhist<<<N_ANCH / 256, 256, 0, stream>>>(labels_pre, noise, ctl, low_fg, 0);
  rpn_k3c<<<1, 256, 0, stream>>>(low_fg, msb_bg, ctl);
  rpn_k3_lowhist<<<N_ANCH / 256, 256, 0, stream>>>(labels_pre, noise, ctl, low_bg, 1);
  rpn_k3e<<<1, 256, 0, stream>>>(low_bg, ctl);
  rpn_k4_final<<<N_ANCH / 256, 256, 0, stream>>>(labels_pre, noise, ctl, out);
}